// SSIMLoss_16020228014220
// MI455X (gfx1250) — compile-verified
//
#include <hip/hip_runtime.h>

typedef __attribute__((ext_vector_type(2))) float v2f;
typedef __attribute__((ext_vector_type(8))) float v8f;

#define HW 512
#define TILE 16
#define TILES_PER_PLANE 1024   /* (512/16)^2 */
#define WAVES_PER_BLOCK 8
#define SSTRIDE 20             /* padded LDS row stride (floats) for 18-wide rows */
#define SSIZE (18 * SSTRIDE)   /* 360 floats per staged image tile */

#define W0 0.30780132f
#define W1 0.38439736f
#define SSIM_C1 1.0e-4f
#define SSIM_C2 9.0e-4f

/* ---- CDNA5 async global->LDS copy (guarded; falls back to sync copy) ---- */
#if defined(__has_builtin)
#if __has_builtin(__builtin_amdgcn_global_load_async_to_lds_b32)
#define HAVE_ASYNC_LDS 1
#endif
#endif

typedef __attribute__((address_space(1))) int as1_int;
typedef __attribute__((address_space(3))) int as3_int;

__device__ __forceinline__ void async_copy_b32(const float* g, float* l) {
#ifdef HAVE_ASYNC_LDS
  __builtin_amdgcn_global_load_async_to_lds_b32(
      (as1_int*)(unsigned long long)g,
      (as3_int*)(unsigned)(unsigned long long)l,
      0, 0);
#else
  *l = *g;
#endif
}

__device__ __forceinline__ void wait_async_copies() {
#ifdef HAVE_ASYNC_LDS
#if __has_builtin(__builtin_amdgcn_s_wait_asynccnt)
  __builtin_amdgcn_s_wait_asynccnt(0);
#else
  asm volatile("s_wait_asynccnt 0" ::: "memory");
#endif
#endif
  asm volatile("" ::: "memory"); /* keep LDS reads below the wait */
}

/* Banded 16x20 Gaussian column-conv matrix: G[m][k] = w(k-m), k-m in {0,1,2} */
__device__ __forceinline__ float gband(int d) {
  return (d == 0) ? W0 : (d == 1) ? W1 : (d == 2) ? W0 : 0.0f;
}

__global__ void __launch_bounds__(256) ssim_zero_ws(float* ws) {
  if (threadIdx.x == 0) ws[0] = 0.0f;
}

__global__ void __launch_bounds__(256) ssim_finalize(float* out, const float* ws) {
  if (threadIdx.x == 0) out[0] = 1.0f - ws[0];
}

__global__ void __launch_bounds__(256)
ssim_main(const float* __restrict__ img1, const float* __restrict__ img2,
          float* __restrict__ ws, int totalWaves) {
  __shared__ float smem[WAVES_PER_BLOCK][2 * SSIZE];
  __shared__ float wsum[WAVES_PER_BLOCK];

  const int lane = threadIdx.x & 31;
  const int wid  = threadIdx.x >> 5;
  const int waveId = blockIdx.x * WAVES_PER_BLOCK + wid;

  float mysum = 0.0f;

  if (waveId < totalWaves) {
    const int plane = waveId >> 10;                  /* n*3 + c */
    const int t     = waveId & (TILES_PER_PLANE - 1);
    const int ty    = (t >> 5) * TILE;
    const int tx    = (t & 31) * TILE;
    const float* p1 = img1 + (size_t)plane * (HW * HW);
    const float* p2 = img2 + (size_t)plane * (HW * HW);

    float* S1 = &smem[wid][0];
    float* S2 = &smem[wid][SSIZE];

    /* ---- stage 18x18 halo tiles; OOB positions get zeros (zero padding) ---- */
    for (int e = lane; e < 18 * 18; e += 32) {
      const int r  = e / 18;
      const int c  = e - r * 18;
      const int gy = ty + r - 1;
      const int gx = tx + c - 1;
      float* d1 = S1 + r * SSTRIDE + c;
      float* d2 = S2 + r * SSTRIDE + c;
      if ((unsigned)gy < (unsigned)HW && (unsigned)gx < (unsigned)HW) {
        const int off = gy * HW + gx;
        async_copy_b32(p1 + off, d1);
        async_copy_b32(p2 + off, d2);
      } else {
        *d1 = 0.0f;
        *d2 = 0.0f;
      }
    }
    wait_async_copies();

    const int m    = lane & 15; /* A: output row M ; B: output column N */
    const int half = lane >> 4;

    v8f aMu1 = {}, aMu2 = {}, aXX = {}, aYY = {}, aXY = {};

    /* Column conv as Out = G(16x20) x R(20x16), 5 chained K=4 WMMAs per quantity.
       Row-conv values R (all 5 quantities) are built per-lane directly into the
       B operand layout: VGPR0 holds K=k0 (lanes 0-15) / K=k0+2 (lanes 16-31). */
#pragma unroll
    for (int chunk = 0; chunk < 5; ++chunk) {
      const int k0 = chunk * 4 + 2 * half;

      v2f A;
      A.x = gband(k0 - m);
      A.y = gband(k0 + 1 - m);

      float b0[5], b1[5];
#pragma unroll
      for (int rr = 0; rr < 2; ++rr) {
        const int r    = k0 + rr;        /* staged row index 0..19 (18,19 = pad) */
        const int rc   = (r < 18) ? r : 17;
        const float vd = (r < 18) ? 1.0f : 0.0f;
        const float* q1 = S1 + rc * SSTRIDE + m;
        const float* q2 = S2 + rc * SSTRIDE + m;
        const float x0 = q1[0], x1 = q1[1], x2 = q1[2];
        const float y0 = q2[0], y1 = q2[1], y2 = q2[2];
        float* dst = rr ? b1 : b0;
        dst[0] = (W0 * x0 + W1 * x1 + W0 * x2) * vd;                   /* mu1  */
        dst[1] = (W0 * y0 + W1 * y1 + W0 * y2) * vd;                   /* mu2  */
        dst[2] = (W0 * x0 * x0 + W1 * x1 * x1 + W0 * x2 * x2) * vd;    /* E[x2]*/
        dst[3] = (W0 * y0 * y0 + W1 * y1 * y1 + W0 * y2 * y2) * vd;    /* E[y2]*/
        dst[4] = (W0 * x0 * y0 + W1 * x1 * y1 + W0 * x2 * y2) * vd;    /* E[xy]*/
      }

      v2f B;
      B.x = b0[0]; B.y = b1[0];
      aMu1 = __builtin_amdgcn_wmma_f32_16x16x4_f32(false, A, false, B, (short)0, aMu1, false, false);
      B.x = b0[1]; B.y = b1[1];
      aMu2 = __builtin_amdgcn_wmma_f32_16x16x4_f32(false, A, false, B, (short)0, aMu2, false, false);
      B.x = b0[2]; B.y = b1[2];
      aXX  = __builtin_amdgcn_wmma_f32_16x16x4_f32(false, A, false, B, (short)0, aXX,  false, false);
      B.x = b0[3]; B.y = b1[3];
      aYY  = __builtin_amdgcn_wmma_f32_16x16x4_f32(false, A, false, B, (short)0, aYY,  false, false);
      B.x = b0[4]; B.y = b1[4];
      aXY  = __builtin_amdgcn_wmma_f32_16x16x4_f32(false, A, false, B, (short)0, aXY,  false, false);
    }

    /* ---- SSIM map on the accumulators, local sum (position-agnostic) ---- */
#pragma unroll
    for (int j = 0; j < 8; ++j) {
      const float mu1 = aMu1[j], mu2 = aMu2[j];
      const float mu1sq = mu1 * mu1;
      const float mu2sq = mu2 * mu2;
      const float mu12  = mu1 * mu2;
      const float s1  = aXX[j] - mu1sq;
      const float s2  = aYY[j] - mu2sq;
      const float s12 = aXY[j] - mu12;
      const float num = (2.0f * mu12 + SSIM_C1) * (2.0f * s12 + SSIM_C2);
      const float den = (mu1sq + mu2sq + SSIM_C1) * (s1 + s2 + SSIM_C2);
      mysum += num / den;
    }
  }

  /* ---- wave32 + block reduction, one atomic per block ---- */
#pragma unroll
  for (int off = 16; off > 0; off >>= 1)
    mysum += __shfl_down(mysum, off, 32);
  if (lane == 0) wsum[wid] = mysum;
  __syncthreads();
  if (threadIdx.x == 0) {
    float bs = 0.0f;
#pragma unroll
    for (int i = 0; i < WAVES_PER_BLOCK; ++i) bs += wsum[i];
    atomicAdd(ws, bs);
  }
}

extern "C" void kernel_launch(void* const* d_in, const int* in_sizes, int n_in,
                              void* d_out, int out_size, void* d_ws, size_t ws_size,
                              hipStream_t stream) {
  const float* img1 = (const float*)d_in[0];
  const float* img2 = (const float*)d_in[1];
  float* out = (float*)d_out;
  float* ws  = (float*)d_ws;

  const int planes     = in_sizes[0] / (HW * HW); /* 32*3 = 96 */
  const int totalWaves = planes * TILES_PER_PLANE;
  const int blocks     = (totalWaves + WAVES_PER_BLOCK - 1) / WAVES_PER_BLOCK;

  ssim_zero_ws<<<1, 256, 0, stream>>>(ws);
  ssim_main<<<blocks, WAVES_PER_BLOCK * 32, 0, stream>>>(img1, img2, ws, totalWaves);
  ssim_finalize<<<1, 256, 0, stream>>>(out, ws);
}